// MSSIMLoss_37580963840053
// MI455X (gfx1250) — compile-verified
//
#include <hip/hip_runtime.h>
#include <math.h>

typedef float v2f __attribute__((ext_vector_type(2)));
typedef float v8f __attribute__((ext_vector_type(8)));

#define C1V 6.5025f
#define C2V 58.5225f

// Gaussian(sigma=1.5), 11 taps, normalized; 0 outside [0,10]
static __device__ __forceinline__ float gw(int t) {
    float r = 0.f;
    r = (t == 0 || t == 10) ? 0.00102838f : r;
    r = (t == 1 || t ==  9) ? 0.00759876f : r;
    r = (t == 2 || t ==  8) ? 0.03600077f : r;
    r = (t == 3 || t ==  7) ? 0.10936073f : r;
    r = (t == 4 || t ==  6) ? 0.21300553f : r;
    r = (t == 5)            ? 0.26601174f : r;
    return r;
}

static __device__ __forceinline__ v8f wmma4(v2f a, v2f b, v8f c) {
    // D = A(16x4,f32) x B(4x16,f32) + C(16x16,f32)
    return __builtin_amdgcn_wmma_f32_16x16x4_f32(false, a, false, b, (short)0, c,
                                                 false, false);
}

// One block: 128 threads (4 waves), one 32x32 output tile of the SSIM map for one (b,c).
__global__ __launch_bounds__(128) void ssimk(const float* __restrict__ X,
                                             const float* __restrict__ Y,
                                             int H, float* __restrict__ pBase) {
    const int W = H, OH = H - 10;
    __shared__ float sInX[48 * 44];
    __shared__ float sInY[48 * 44];
    __shared__ float sH[5][48 * 33];
    __shared__ float sRed[256];
    __shared__ float sGW[64];   // zero-padded Gaussian: sGW[15+t] = g[t]

    const int tid  = threadIdx.x;
    const int lane = tid & 31;
    const int wids = __builtin_amdgcn_readfirstlane(tid >> 5);  // scalar wave id
    const int mlo  = lane & 15;
    const int ksel = lane >> 4;

    const int bc = blockIdx.z;
    const int y0 = blockIdx.y * 32;
    const int x0 = blockIdx.x * 32;
    const size_t base = (size_t)bc * H * W;

    if (tid < 64) sGW[tid] = gw(tid - 15);

    // Load 42x42 region (zero-padded to 48x44) of X and Y into LDS.
    {
        int r = tid / 44;
        int cc = tid - r * 44;
        for (int i = tid; i < 48 * 44; i += 128) {
            float vx = 0.f, vy = 0.f;
            if (r < 42 && cc < 42) {
                int gy = min(y0 + r, H - 1);
                int gx = min(x0 + cc, W - 1);
                size_t o = base + (size_t)gy * W + gx;
                vx = X[o];
                vy = Y[o];
            }
            sInX[i] = vx;
            sInY[i] = vy;
            r += 2; cc += 40;                  // advance by 128 = 2*44 + 40
            if (cc >= 44) { cc -= 44; r += 1; }
        }
    }

    // Per-lane Gaussian fragment base: wf(j) = { g[4j+2k-mlo], g[4j+2k+1-mlo] }
    // (B operand of horizontal pass, A operand of vertical pass).
    const int bgw = 15 + 2 * ksel - mlo;   // in [0,17]; +4j+1 stays < 64
    auto wfj = [&](int j) -> v2f {
        return (v2f){sGW[bgw + 4 * j], sGW[bgw + 4 * j + 1]};
    };
    __syncthreads();

    // ---- Horizontal pass: 6 tile-tasks (3 M-tiles x 2 N-tiles), 5 quantities each ----
    auto htile = [&](int t) {
        const int R0 = (t >> 1) * 16;   // scalar
        const int nb = (t & 1) * 16;    // scalar
        const int rb = (R0 + mlo) * 44 + nb + 2 * ksel;
        v8f aX  = {0.f, 0.f, 0.f, 0.f, 0.f, 0.f, 0.f, 0.f};
        v8f aY  = aX, aXX = aX, aYY = aX, aXY = aX;
#pragma unroll
        for (int j = 0; j < 7; j++) {
            int ro = rb + 4 * j;
            float ax0 = sInX[ro], ax1 = sInX[ro + 1];
            float ay0 = sInY[ro], ay1 = sInY[ro + 1];
            v2f w = wfj(j);
            aX  = wmma4((v2f){ax0, ax1}, w, aX);
            aY  = wmma4((v2f){ay0, ay1}, w, aY);
            aXX = wmma4((v2f){ax0 * ax0, ax1 * ax1}, w, aXX);
            aYY = wmma4((v2f){ay0 * ay0, ay1 * ay1}, w, aYY);
            aXY = wmma4((v2f){ax0 * ay0, ax1 * ay1}, w, aXY);
        }
        const int sb = (R0 + 8 * ksel) * 33 + nb + mlo;
#pragma unroll
        for (int v = 0; v < 8; v++) {
            sH[0][sb + v * 33] = aX[v];
            sH[1][sb + v * 33] = aY[v];
            sH[2][sb + v * 33] = aXX[v];
            sH[3][sb + v * 33] = aYY[v];
            sH[4][sb + v * 33] = aXY[v];
        }
    };
    htile(wids);                       // tasks 0..3
    if (wids < 2) htile(wids + 4);     // tasks 4..5 (scalar branch)
    __syncthreads();

    // ---- Vertical pass: one 16x16 output tile per wave, all 5 quantities ----
    const int yb = (wids >> 1) * 16;   // scalar
    const int xb = (wids & 1) * 16;    // scalar
    v8f am0 = {0.f, 0.f, 0.f, 0.f, 0.f, 0.f, 0.f, 0.f};
    v8f am1 = am0, am2 = am0, am3 = am0, am4 = am0;
    const int vcol = xb + mlo;
#pragma unroll
    for (int j = 0; j < 7; j++) {
        int r0 = (yb + 4 * j + 2 * ksel) * 33 + vcol;
        v2f w = wfj(j);
        am0 = wmma4(w, (v2f){sH[0][r0], sH[0][r0 + 33]}, am0);
        am1 = wmma4(w, (v2f){sH[1][r0], sH[1][r0 + 33]}, am1);
        am2 = wmma4(w, (v2f){sH[2][r0], sH[2][r0 + 33]}, am2);
        am3 = wmma4(w, (v2f){sH[3][r0], sH[3][r0 + 33]}, am3);
        am4 = wmma4(w, (v2f){sH[4][r0], sH[4][r0 + 33]}, am4);
    }

    // ---- Per-pixel SSIM / CS + masked accumulation ----
    float sS = 0.f, sC = 0.f;
#pragma unroll
    for (int v = 0; v < 8; v++) {
        int oy = y0 + yb + v + 8 * ksel;
        int ox = x0 + xb + mlo;
        float mu1 = am0[v], mu2 = am1[v];
        float exx = am2[v], eyy = am3[v], exy = am4[v];
        float m11 = mu1 * mu1, m22 = mu2 * mu2, m12 = mu1 * mu2;
        float s1 = exx - m11, s2 = eyy - m22, s12 = exy - m12;
        float cs = (2.f * s12 + C2V) / (s1 + s2 + C2V);
        float ss = ((2.f * m12 + C1V) / (m11 + m22 + C1V)) * cs;
        if (oy < OH && ox < OH) { sS += ss; sC += cs; }
    }

    sRed[tid] = sS;
    sRed[128 + tid] = sC;
    __syncthreads();
    for (int s = 64; s > 0; s >>= 1) {
        if (tid < s) {
            sRed[tid] += sRed[tid + s];
            sRed[128 + tid] += sRed[128 + tid + s];
        }
        __syncthreads();
    }
    if (tid == 0) {
        int ntiles = gridDim.x * gridDim.y;
        int tile = blockIdx.y * gridDim.x + blockIdx.x;
        pBase[bc * ntiles + tile] = sRed[0];                 // ssim partial
        pBase[96 * ntiles + bc * ntiles + tile] = sRed[128]; // cs partial
    }
}

// 2x2 average pool (stride 2) for X and Y simultaneously. Hn = output side.
__global__ void dsk(const float* __restrict__ inX, const float* __restrict__ inY,
                    float* __restrict__ outX, float* __restrict__ outY, int Hn) {
    int idx = blockIdx.x * blockDim.x + threadIdx.x;
    int plane = Hn * Hn;
    if (idx >= 96 * plane) return;
    int bc = idx / plane, r = idx % plane;
    int y = r / Hn, x = r % Hn;
    int Hi = 2 * Hn;
    size_t o = (size_t)bc * Hi * Hi + (size_t)(2 * y) * Hi + 2 * x;
    outX[idx] = 0.25f * (inX[o] + inX[o + 1] + inX[o + Hi] + inX[o + Hi + 1]);
    outY[idx] = 0.25f * (inY[o] + inY[o + 1] + inY[o + Hi] + inY[o + Hi + 1]);
}

// Fixed-order tile reduction + weighted power product + mean -> 1 - ms_ssim.
__global__ __launch_bounds__(128) void finalk(const float* __restrict__ part,
                                              float* __restrict__ out) {
    __shared__ float red[128];
    const int tid = threadIdx.x;
    const int   NT[5]  = {144, 36, 9, 4, 1};
    const int   OFF[5] = {0, 27648, 34560, 36288, 37056};
    const float NP[5]  = {139876.f, 33124.f, 7396.f, 1444.f, 196.f};
    const float WG[5]  = {0.0448f, 0.2856f, 0.3001f, 0.2363f, 0.1333f};
    float prod = 0.f;
    if (tid < 96) {
        prod = 1.f;
        for (int s = 0; s < 5; s++) {
            const float* pb = part + OFF[s];
            int nt = NT[s];
            float sumS = 0.f, sumC = 0.f;
            for (int t = 0; t < nt; t++) {
                sumS += pb[tid * nt + t];
                sumC += pb[96 * nt + tid * nt + t];
            }
            float val = (s < 4) ? (sumC / NP[s]) : (sumS / NP[s]);
            val = fmaxf(val, 0.f);
            prod *= powf(val, WG[s]);
        }
    }
    red[tid] = prod;
    __syncthreads();
    for (int s = 64; s > 0; s >>= 1) {
        if (tid < s) red[tid] += red[tid + s];
        __syncthreads();
    }
    if (tid == 0) out[0] = 1.f - red[0] / 96.f;
}

extern "C" void kernel_launch(void* const* d_in, const int* in_sizes, int n_in,
                              void* d_out, int out_size, void* d_ws, size_t ws_size,
                              hipStream_t stream) {
    (void)in_sizes; (void)n_in; (void)out_size; (void)ws_size;
    const float* X0 = (const float*)d_in[0];
    const float* Y0 = (const float*)d_in[1];
    float* ws = (float*)d_ws;

    const size_t n1 = 96UL * 192 * 192;
    const size_t n2 = 96UL * 96 * 96;
    const size_t n3 = 96UL * 48 * 48;
    const size_t n4 = 96UL * 24 * 24;
    float* X1 = ws;            float* Y1 = X1 + n1;
    float* X2 = Y1 + n1;       float* Y2 = X2 + n2;
    float* X3 = Y2 + n2;       float* Y3 = X3 + n3;
    float* X4 = Y3 + n3;       float* Y4 = X4 + n4;
    float* part = Y4 + n4;     // 37248 floats of tile partials

    // Scale 0 (384)
    ssimk<<<dim3(12, 12, 96), 128, 0, stream>>>(X0, Y0, 384, part + 0);
    dsk<<<(int)((n1 + 255) / 256), 256, 0, stream>>>(X0, Y0, X1, Y1, 192);
    // Scale 1 (192)
    ssimk<<<dim3(6, 6, 96), 128, 0, stream>>>(X1, Y1, 192, part + 27648);
    dsk<<<(int)((n2 + 255) / 256), 256, 0, stream>>>(X1, Y1, X2, Y2, 96);
    // Scale 2 (96)
    ssimk<<<dim3(3, 3, 96), 128, 0, stream>>>(X2, Y2, 96, part + 34560);
    dsk<<<(int)((n3 + 255) / 256), 256, 0, stream>>>(X2, Y2, X3, Y3, 48);
    // Scale 3 (48)
    ssimk<<<dim3(2, 2, 96), 128, 0, stream>>>(X3, Y3, 48, part + 36288);
    dsk<<<(int)((n4 + 255) / 256), 256, 0, stream>>>(X3, Y3, X4, Y4, 24);
    // Scale 4 (24)
    ssimk<<<dim3(1, 1, 96), 128, 0, stream>>>(X4, Y4, 24, part + 37056);

    finalk<<<1, 128, 0, stream>>>(part, (float*)d_out);
}